// LightGCN_28415503630349
// MI455X (gfx1250) — compile-verified
//
#include <hip/hip_runtime.h>
#include <hip/hip_bf16.h>

#define NUM_USERS 40000
#define NUM_ITEMS 60000
#define N_NODES   100000
#define DIM       64
#define EPS_F     1e-7f
#define MIN_NORM_F 1e-15f

typedef __attribute__((ext_vector_type(2))) float v2f;
typedef __attribute__((ext_vector_type(8))) float v8f;

// ---------------------------------------------------------------------------
// Preprocess: X[n] = hyp_logmap0(hyp_proj(w[n])) for C=1.
//   s      = sum_{d>=1} w[n][d]^2        (computed on the WMMA pipe)
//   x0     = sqrt(max(1+s, EPS)); theta = max(x0, 1+EPS)
//   arc    = log(theta + sqrt(theta^2-1)); scale = arc / max(sqrt(s),1e-15)
//   X[n][0]=0 ; X[n][d]=scale*w[n][d]
// One wave handles 16 nodes. Sum-of-squares via V_WMMA_F32_16X16X4_F32 with a
// ones B matrix: D[m][n] = rowsum(A[m][:]) for every n, accumulated over 16
// K=4 chunks covering the 64 dims (dim 0 masked to zero in chunk 0).
// ---------------------------------------------------------------------------
__global__ __launch_bounds__(256) void prep_kernel(const float* __restrict__ uw,
                                                   const float* __restrict__ iw,
                                                   float* __restrict__ X) {
  const int wave = (int)((blockIdx.x * blockDim.x + threadIdx.x) >> 5);
  const int lane = threadIdx.x & 31;
  const int base = wave * 16;                  // first node of this wave
  if (base >= N_NODES) return;                 // wave-uniform: EXEC stays full
  // NUM_USERS is a multiple of 16, so each wave's 16 nodes live in one array.
  const float* W = (base < NUM_USERS) ? (uw + (size_t)base * DIM)
                                      : (iw + (size_t)(base - NUM_USERS) * DIM);
  // A-matrix layout for 32-bit 16x4: lanes 0-15 hold K=0,1; lanes 16-31 K=2,3.
  const int m = lane & 15;
  const float* rowm = W + (size_t)m * DIM;
  const int koff = (lane < 16) ? 0 : 2;

  v8f acc = {0.f, 0.f, 0.f, 0.f, 0.f, 0.f, 0.f, 0.f};
  v2f ones; ones[0] = 1.0f; ones[1] = 1.0f;    // B = all-ones -> D = rowsums(A)

#pragma unroll
  for (int k = 0; k < 16; ++k) {
    float a0 = rowm[4 * k + koff];
    float a1 = rowm[4 * k + koff + 1];
    if (k == 0) a0 = (lane < 16) ? 0.0f : a0;  // exclude dim 0 (cndmask, no branch)
    v2f A; A[0] = a0 * a0; A[1] = a1 * a1;
    acc = __builtin_amdgcn_wmma_f32_16x16x4_f32(false, A, false, ones,
                                                (short)0, acc, false, false);
  }

  // D layout: VGPR r = {M=r in lanes 0-15, M=r+8 in lanes 16-31}, replicated
  // over N. Broadcast each node's sum to all lanes and write its scaled row.
  float* Xw = X + (size_t)base * DIM;
#pragma unroll
  for (int j = 0; j < 16; ++j) {
    const float s  = __shfl(acc[j & 7], (j >> 3) * 16, 32);
    const float x0 = sqrtf(fmaxf(1.0f + s, EPS_F));
    const float th = fmaxf(x0, 1.0f + EPS_F);
    const float arc = logf(th + sqrtf(th * th - 1.0f));
    const float yn  = fmaxf(sqrtf(s), MIN_NORM_F);
    const float scale = arc / yn;
    const float2 wv = *(const float2*)(W + (size_t)j * DIM + 2 * lane);
    float2 ov;
    ov.x = (lane == 0) ? 0.0f : scale * wv.x;  // X[n][0] = 0
    ov.y = scale * wv.y;
    *(float2*)(Xw + (size_t)j * DIM + 2 * lane) = ov;                 // 256B/row
  }
}

// ---------------------------------------------------------------------------
// SpMM: y[row] += val * x[col].  One wave32 per edge, float2 per lane:
// one fully-coalesced 256B gather + 64 fire-and-forget global_atomic_add_f32.
// ---------------------------------------------------------------------------
__global__ __launch_bounds__(256) void spmm_kernel(const int* __restrict__ rows,
                                                   const int* __restrict__ cols,
                                                   const float* __restrict__ vals,
                                                   const float* __restrict__ x,
                                                   float* __restrict__ y, int nnz) {
  const long long tid = (long long)blockIdx.x * blockDim.x + threadIdx.x;
  const int e = (int)(tid >> 5);
  if (e >= nnz) return;
  const int lane = (int)(tid & 31);
  const int r = rows[e];
  const int c = cols[e];
  const float v = vals[e];
  const float2 xv = *(const float2*)(x + (size_t)c * DIM + 2 * lane);
  float* dst = y + (size_t)r * DIM + 2 * lane;
  atomicAdd(dst + 0, v * xv.x);   // no-return -> STOREcnt only
  atomicAdd(dst + 1, v * xv.y);
}

__global__ __launch_bounds__(256) void zero_kernel(float4* __restrict__ p, long long n4) {
  const long long i = (long long)blockIdx.x * blockDim.x + threadIdx.x;
  if (i < n4) p[i] = make_float4(0.f, 0.f, 0.f, 0.f);
}

// light += y; optionally zero z (the buffer that becomes the next SpMM target)
__global__ __launch_bounds__(256) void acc_kernel(float4* __restrict__ light,
                                                  const float4* __restrict__ y,
                                                  float4* __restrict__ z, long long n4) {
  const long long i = (long long)blockIdx.x * blockDim.x + threadIdx.x;
  if (i >= n4) return;
  float4 l = light[i];
  const float4 a = y[i];
  l.x += a.x; l.y += a.y; l.z += a.z; l.w += a.w;
  light[i] = l;
  if (z) z[i] = make_float4(0.f, 0.f, 0.f, 0.f);
}

// out = [ light[user[i]] (4096x64) | light[NUM_USERS+pos[i]] (4096x64) | items ]
__global__ __launch_bounds__(256) void gather_kernel(const float* __restrict__ light,
                                                     const int* __restrict__ user,
                                                     const int* __restrict__ pos,
                                                     float* __restrict__ out) {
  const int NU64 = 4096 * DIM;
  const int TOT  = 2 * NU64 + NUM_ITEMS * DIM;
  const int i = (int)(blockIdx.x * blockDim.x + threadIdx.x);
  if (i >= TOT) return;
  if (i < NU64) {
    const int u = user[i >> 6];
    out[i] = light[(size_t)u * DIM + (i & 63)];
  } else if (i < 2 * NU64) {
    const int j = i - NU64;
    const int p = pos[j >> 6];
    out[i] = light[((size_t)NUM_USERS + (size_t)p) * DIM + (j & 63)];
  } else {
    out[i] = light[(size_t)NUM_USERS * DIM + (size_t)(i - 2 * NU64)];
  }
}

extern "C" void kernel_launch(void* const* d_in, const int* in_sizes, int n_in,
                              void* d_out, int out_size, void* d_ws, size_t ws_size,
                              hipStream_t stream) {
  const float* uw   = (const float*)d_in[0];
  const float* iw   = (const float*)d_in[1];
  const float* vals = (const float*)d_in[2];
  const int*   rows = (const int*)d_in[3];
  const int*   cols = (const int*)d_in[4];
  const int*   user = (const int*)d_in[5];
  const int*   pos  = (const int*)d_in[6];
  float* out = (float*)d_out;
  const int nnz = in_sizes[2];

  const long long NN64 = (long long)N_NODES * DIM;   // 6.4M floats (25.6 MB)
  float* A = (float*)d_ws;                           // ping
  float* B = A + NN64;                               // pong
  float* L = B + NN64;                               // light_out accumulator

  // Zero B and L (contiguous) — ws is poisoned by the harness.
  {
    const long long n4 = (2 * NN64) / 4;
    zero_kernel<<<(int)((n4 + 255) / 256), 256, 0, stream>>>((float4*)B, n4);
  }

  // Preprocess into A: 6250 waves (16 nodes each).
  {
    const int threads = (N_NODES / 16) * 32;
    prep_kernel<<<(threads + 255) / 256, 256, 0, stream>>>(uw, iw, A);
  }

  const long long spmmThreads = (long long)nnz * 32;
  const int spmmBlocks = (int)((spmmThreads + 255) / 256);
  const long long n4 = NN64 / 4;
  const int accBlocks = (int)((n4 + 255) / 256);

  // Layer 1: B = A*Adj ; L += B ; zero A (next target)
  spmm_kernel<<<spmmBlocks, 256, 0, stream>>>(rows, cols, vals, A, B, nnz);
  acc_kernel<<<accBlocks, 256, 0, stream>>>((float4*)L, (const float4*)B, (float4*)A, n4);
  // Layer 2: A = B*Adj ; L += A ; zero B
  spmm_kernel<<<spmmBlocks, 256, 0, stream>>>(rows, cols, vals, B, A, nnz);
  acc_kernel<<<accBlocks, 256, 0, stream>>>((float4*)L, (const float4*)A, (float4*)B, n4);
  // Layer 3: B = A*Adj ; L += B
  spmm_kernel<<<spmmBlocks, 256, 0, stream>>>(rows, cols, vals, A, B, nnz);
  acc_kernel<<<accBlocks, 256, 0, stream>>>((float4*)L, (const float4*)B, (float4*)nullptr, n4);

  // Gathers + items copy.
  {
    const int TOT = 2 * 4096 * DIM + NUM_ITEMS * DIM;
    gather_kernel<<<(TOT + 255) / 256, 256, 0, stream>>>(L, user, pos, out);
  }
}